// Decoder_23991687315866
// MI455X (gfx1250) — compile-verified
//
#include <hip/hip_runtime.h>
#include <hip/hip_bf16.h>

typedef float v2f __attribute__((ext_vector_type(2)));
typedef float v8f __attribute__((ext_vector_type(8)));

#define BB   4096   // batch
#define DD   32     // dec_x feature dim
#define HH   5      // hidden dim
#define GG   20     // 4*H gates
#define INW  128    // input width = D + 3*E

__device__ __forceinline__ float sigf(float x) { return 1.0f / (1.0f + __expf(-x)); }

// ---------------------------------------------------------------------------
// Precompute per-b gate biases: hb[l][b][g] = hidden[l][b] @ W_hh[l].T + b_ih + b_hh
// ---------------------------------------------------------------------------
__global__ __launch_bounds__(256) void hb_kernel(
    const float* __restrict__ hidden,
    const float* __restrict__ W_hh0, const float* __restrict__ b_ih0, const float* __restrict__ b_hh0,
    const float* __restrict__ W_hh1, const float* __restrict__ b_ih1, const float* __restrict__ b_hh1,
    float* __restrict__ hb0, float* __restrict__ hb1)
{
    int b = blockIdx.x * blockDim.x + threadIdx.x;
    if (b >= BB) return;
    float h0[HH], h1[HH];
    #pragma unroll
    for (int j = 0; j < HH; ++j) {
        h0[j] = hidden[b * HH + j];
        h1[j] = hidden[BB * HH + b * HH + j];
    }
    #pragma unroll 4
    for (int g = 0; g < GG; ++g) {
        float s0 = b_ih0[g] + b_hh0[g];
        float s1 = b_ih1[g] + b_hh1[g];
        #pragma unroll
        for (int j = 0; j < HH; ++j) {
            s0 += h0[j] * W_hh0[g * HH + j];
            s1 += h1[j] * W_hh1[g * HH + j];
        }
        hb0[b * GG + g] = s0;
        hb1[b * GG + g] = s1;
    }
}

// ---------------------------------------------------------------------------
// Main kernel: 1 wave = 16 rows (t fixed, consecutive b).
// Layer-0 gates via V_WMMA_F32_16X16X4_F32 (K=128 in 32 steps, two N-tiles).
// ---------------------------------------------------------------------------
__global__ __launch_bounds__(256) void lstm_wmma_kernel(
    const float* __restrict__ dec_x,
    const int*   __restrict__ cat1,
    const int*   __restrict__ cat2,
    const int*   __restrict__ cat3,
    const float* __restrict__ emb,
    const float* __restrict__ W_ih0,
    const float* __restrict__ W_ih1,
    const float* __restrict__ cell,
    const float* __restrict__ hb0,
    const float* __restrict__ hb1,
    float* __restrict__ out)
{
    __shared__ float Wp[32 * INW];       // zero-padded W_ih0: rows 0-19 real, 20-31 zero
    __shared__ float Gs[8][16 * GG];     // per-wave gate tile [row][gate]

    const int tid = threadIdx.x;
    // Cooperative load of padded weights into LDS.
    for (int i = tid; i < 32 * INW; i += 256) {
        Wp[i] = ((i >> 7) < GG) ? W_ih0[i] : 0.0f;
    }
    __syncthreads();

    const int wave = tid >> 5;
    const int lane = tid & 31;
    const int m    = lane & 15;          // row within tile / N col for B-frag
    const int half = lane >> 4;
    const int koff = half * 2;           // K sub-offset per ISA A/B layout

    const long rbase = ((long)blockIdx.x * 8 + wave) * 16;   // 16 | 4096 -> same t per wave
    const int  t     = (int)(rbase >> 12);
    const int  b     = (int)(rbase & 4095) + m;
    const long row   = (long)t * BB + b;

    // Four 32-float segments of the concatenated x row.
    const float* srcs[4];
    srcs[0] = dec_x + row * DD;
    srcs[1] = emb + (long)cat1[row] * 32;
    srcs[2] = emb + (long)cat2[row] * 32;
    srcs[3] = emb + (long)cat3[row] * 32;

    const float* bp0 = &Wp[m * INW];          // B-frag rows for N-tile 0 (gates 0-15)
    const float* bp1 = &Wp[(m + 16) * INW];   // B-frag rows for N-tile 1 (gates 16-19 + pad)

    v8f acc0 = {};
    v8f acc1 = {};
    #pragma unroll
    for (int seg = 0; seg < 4; ++seg) {
        const float* ap = srcs[seg];
        #pragma unroll
        for (int kk = 0; kk < 8; ++kk) {
            const int col  = kk * 4 + koff;       // column within segment
            const int gcol = seg * 32 + col;      // global K column
            v2f a  = *(const v2f*)(ap  + col);
            v2f b0 = *(const v2f*)(bp0 + gcol);
            v2f b1 = *(const v2f*)(bp1 + gcol);
            acc0 = __builtin_amdgcn_wmma_f32_16x16x4_f32(false, a, false, b0,
                                                         (short)0, acc0, false, false);
            acc1 = __builtin_amdgcn_wmma_f32_16x16x4_f32(false, a, false, b1,
                                                         (short)0, acc1, false, false);
        }
    }

    // Spill D tiles to per-wave LDS: VGPR j holds rows (j, j+8) per C/D layout.
    float* gs = &Gs[wave][0];
    #pragma unroll
    for (int j = 0; j < 8; ++j) {
        const int rm = j + half * 8;
        gs[rm * GG + m] = acc0[j];
        if (m < 4) gs[rm * GG + 16 + m] = acc1[j];
    }
    asm volatile("s_wait_dscnt 0" ::: "memory");

    // Lanes 0-15: one row each -> elementwise LSTM + tiny layer-1 GEMV.
    if (lane < 16) {
        float g0[GG];
        #pragma unroll
        for (int g = 0; g < GG; ++g) g0[g] = gs[m * GG + g] + hb0[(long)b * GG + g];

        float h1v[HH];
        #pragma unroll
        for (int h = 0; h < HH; ++h) {
            float cn = sigf(g0[5 + h]) * cell[(long)b * HH + h]
                     + sigf(g0[h]) * tanhf(g0[10 + h]);
            h1v[h] = sigf(g0[15 + h]) * tanhf(cn);
        }

        float g1[GG];
        #pragma unroll 4
        for (int g = 0; g < GG; ++g) {
            float s = hb1[(long)b * GG + g];
            #pragma unroll
            for (int j = 0; j < HH; ++j) s += W_ih1[g * HH + j] * h1v[j];
            g1[g] = s;
        }
        #pragma unroll
        for (int h = 0; h < HH; ++h) {
            float cn = sigf(g1[5 + h]) * cell[(long)BB * HH + (long)b * HH + h]
                     + sigf(g1[h]) * tanhf(g1[10 + h]);
            out[row * HH + h] = sigf(g1[15 + h]) * tanhf(cn);
        }
    }
}

// ---------------------------------------------------------------------------
extern "C" void kernel_launch(void* const* d_in, const int* in_sizes, int n_in,
                              void* d_out, int out_size, void* d_ws, size_t ws_size,
                              hipStream_t stream)
{
    const float* hidden = (const float*)d_in[1];
    const float* cellp  = (const float*)d_in[2];
    const float* dec_x  = (const float*)d_in[3];
    const int*   cat1   = (const int*)d_in[4];
    const int*   cat2   = (const int*)d_in[5];
    const int*   cat3   = (const int*)d_in[6];
    const float* emb    = (const float*)d_in[7];
    const float* W_ih0  = (const float*)d_in[8];
    const float* W_hh0  = (const float*)d_in[9];
    const float* b_ih0  = (const float*)d_in[10];
    const float* b_hh0  = (const float*)d_in[11];
    const float* W_ih1  = (const float*)d_in[12];
    const float* W_hh1  = (const float*)d_in[13];
    const float* b_ih1  = (const float*)d_in[14];
    const float* b_hh1  = (const float*)d_in[15];
    float* out = (float*)d_out;

    const int T    = in_sizes[3] / (BB * DD);   // = horizon = 64
    const int rows = T * BB;

    float* hb0 = (float*)d_ws;
    float* hb1 = hb0 + (size_t)BB * GG;

    hb_kernel<<<BB / 256, 256, 0, stream>>>(hidden, W_hh0, b_ih0, b_hh0,
                                            W_hh1, b_ih1, b_hh1, hb0, hb1);

    lstm_wmma_kernel<<<rows / 128, 256, 0, stream>>>(dec_x, cat1, cat2, cat3, emb,
                                                     W_ih0, W_ih1, cellp,
                                                     hb0, hb1, out);
}